// GATLayer_77326591197513
// MI455X (gfx1250) — compile-verified
//
#include <hip/hip_runtime.h>
#include <hip/hip_bf16.h>

typedef float v2f __attribute__((ext_vector_type(2)));
typedef float v8f __attribute__((ext_vector_type(8)));

#define IN_DIM   64
#define OUT_DIM  64
#define EDGE_DIM 16
#define H_HEADS  4
#define HD       16
#define NEG_SLOPE   0.2f
#define EPS_SOFTMAX 1e-8f
#define EPS_LN      1e-5f

// ---------------------------------------------------------------- zero scratch
__global__ void zero_kernel(float* __restrict__ p, long n) {
    long i = (long)blockIdx.x * blockDim.x + threadIdx.x;
    long stride = (long)gridDim.x * blockDim.x;
    for (; i < n; i += stride) p[i] = 0.0f;
}

// ---------------------------------------------------------------- h = x @ W
// One wave per 16x16 output tile of h. Block = 128 threads = 4 waves covering
// the 4 column tiles (64 cols) of one 16-row band. K=64 accumulated in 16
// V_WMMA_F32_16X16X4_F32 steps.
// A (16x4 f32) per ISA: lane l: M=l&15, Kpair=(l>>4)*2 -> v2f {A[M][K], A[M][K+1]}
// B (4x16 f32) mirrored: lane l: N=l&15, Kpair=(l>>4)*2 -> v2f {B[K][N], B[K+1][N]}
// C/D (16x16 f32): VGPR r, lane l -> row r + 8*(l>>4), col l&15
__global__ void node_gemm_wmma(const float* __restrict__ x,
                               const float* __restrict__ W,
                               float* __restrict__ h, int n) {
    int wave = threadIdx.x >> 5;        // 0..3 -> column tile
    int lane = threadIdx.x & 31;
    int grp  = lane >> 4;               // 0 or 1
    int lid  = lane & 15;

    int row0 = blockIdx.x * 16;
    int col0 = wave * 16;

    int arow = row0 + lid;
    if (arow >= n) arow = n - 1;        // clamp loads; EXEC stays all-1s

    const float* xr = x + (long)arow * IN_DIM + 2 * grp;          // A row ptr
    const float* wc = W + (long)(2 * grp) * OUT_DIM + col0 + lid; // B col ptr

    v8f c = {};
    #pragma unroll
    for (int k0 = 0; k0 < IN_DIM; k0 += 4) {
        v2f a = *(const v2f*)(xr + k0);     // {x[row][k0+2g], x[row][k0+2g+1]}
        v2f b;
        b.x = wc[(long)k0 * OUT_DIM];       // W[k0+2g  ][col]
        b.y = wc[(long)(k0 + 1) * OUT_DIM]; // W[k0+2g+1][col]
        c = __builtin_amdgcn_wmma_f32_16x16x4_f32(false, a, false, b,
                                                  (short)0, c, false, false);
    }

    #pragma unroll
    for (int r = 0; r < 8; ++r) {
        int row = row0 + r + 8 * grp;
        if (row < n) h[(long)row * OUT_DIM + col0 + lid] = c[r];
    }
}

// ------------------------------------------- per-(node,head) attn projections
__global__ void node_proj(const float* __restrict__ h,
                          const float* __restrict__ attn_src,
                          const float* __restrict__ attn_dst,
                          float* __restrict__ a_src,
                          float* __restrict__ a_dst, int n) {
    int idx = blockIdx.x * blockDim.x + threadIdx.x;
    if (idx >= n * H_HEADS) return;
    int hh   = idx & 3;
    int node = idx >> 2;
    const float4* hp = (const float4*)(h + (long)node * OUT_DIM + hh * HD);
    const float4* as = (const float4*)(attn_src + hh * HD);
    const float4* ad = (const float4*)(attn_dst + hh * HD);
    float s = 0.0f, d = 0.0f;
    #pragma unroll
    for (int j = 0; j < 4; ++j) {
        float4 hv = hp[j], sv = as[j], dv = ad[j];
        s += hv.x * sv.x + hv.y * sv.y + hv.z * sv.z + hv.w * sv.w;
        d += hv.x * dv.x + hv.y * dv.y + hv.z * dv.z + hv.w * dv.w;
    }
    a_src[idx] = s;
    a_dst[idx] = d;
}

// ------------------- edge scores + leaky relu + segment max (int-bit atomic)
__global__ void edge_score(const int* __restrict__ src,
                           const int* __restrict__ dst,
                           const float* __restrict__ edge_attr,
                           const float* __restrict__ W_edge,
                           const float* __restrict__ a_src,
                           const float* __restrict__ a_dst,
                           float* __restrict__ attn,
                           int* __restrict__ amax_bits, long e_cnt) {
    long e = (long)blockIdx.x * blockDim.x + threadIdx.x;
    if (e >= e_cnt) return;
    int s = src[e], t = dst[e];

    float sc[H_HEADS] = {0.f, 0.f, 0.f, 0.f};
    const float4* ea = (const float4*)(edge_attr + e * EDGE_DIM);
    #pragma unroll
    for (int q = 0; q < 4; ++q) {
        float4 v = ea[q];
        int d0 = q * 4;
        #pragma unroll
        for (int hh = 0; hh < H_HEADS; ++hh) {
            sc[hh] += v.x * W_edge[(d0 + 0) * H_HEADS + hh]
                    + v.y * W_edge[(d0 + 1) * H_HEADS + hh]
                    + v.z * W_edge[(d0 + 2) * H_HEADS + hh]
                    + v.w * W_edge[(d0 + 3) * H_HEADS + hh];
        }
    }
    float4 asv = ((const float4*)a_src)[s];
    float4 adv = ((const float4*)a_dst)[t];
    float av[H_HEADS] = {asv.x + adv.x, asv.y + adv.y, asv.z + adv.z, asv.w + adv.w};

    float4 outv;
    float* op = (float*)&outv;
    #pragma unroll
    for (int hh = 0; hh < H_HEADS; ++hh) {
        float v = av[hh] + sc[hh];
        v = (v >= 0.0f) ? v : NEG_SLOPE * v;
        op[hh] = v;
        // amax starts at 0.0 and is only ever >= 0; for non-negative IEEE
        // floats integer ordering == float ordering -> native atomic_max_i32.
        if (v > 0.0f) atomicMax(&amax_bits[t * H_HEADS + hh], __float_as_int(v));
    }
    ((float4*)attn)[e] = outv;
}

// ---------------------------------- exp(attn - max[dst]) + segment sum
__global__ void edge_softmax(const int* __restrict__ dst,
                             const float* __restrict__ amax,
                             float* __restrict__ attn,
                             float* __restrict__ asum, long eh) {
    long idx = (long)blockIdx.x * blockDim.x + threadIdx.x;
    if (idx >= eh) return;
    long e = idx >> 2;
    int hh = (int)(idx & 3);
    int t  = dst[e];
    float v = expf(attn[idx] - amax[t * H_HEADS + hh]);
    attn[idx] = v;
    atomicAdd(&asum[t * H_HEADS + hh], v);
}

// ---------------------------------- normalized message scatter-add
__global__ void scatter_msg(const int* __restrict__ src,
                            const int* __restrict__ dst,
                            const float* __restrict__ h,
                            const float* __restrict__ attn,
                            const float* __restrict__ asum,
                            float* __restrict__ acc, long total) {
    long idx = (long)blockIdx.x * blockDim.x + threadIdx.x;
    if (idx >= total) return;
    long e = idx >> 6;
    int d  = (int)(idx & 63);
    int hh = d >> 4;
    int s = src[e], t = dst[e];
    float w = attn[e * H_HEADS + hh] / (asum[t * H_HEADS + hh] + EPS_SOFTMAX);
    atomicAdd(&acc[(long)t * OUT_DIM + d], h[(long)s * OUT_DIM + d] * w);
}

// ---------------------------------- residual + layernorm, one wave32 per node
__global__ void residual_ln(const float* __restrict__ acc,
                            const float* __restrict__ h,
                            const float* __restrict__ gamma,
                            const float* __restrict__ beta,
                            float* __restrict__ out, int n) {
    int wave = threadIdx.x >> 5;
    int lane = threadIdx.x & 31;
    int node = blockIdx.x * 8 + wave;
    if (node >= n) return;
    long base = (long)node * OUT_DIM;

    float y0 = acc[base + lane]      + h[base + lane];
    float y1 = acc[base + lane + 32] + h[base + lane + 32];

    float s = y0 + y1;
    #pragma unroll
    for (int off = 16; off > 0; off >>= 1) s += __shfl_xor(s, off, 32);
    float mu = s * (1.0f / 64.0f);

    float d0 = y0 - mu, d1 = y1 - mu;
    float vs = d0 * d0 + d1 * d1;
    #pragma unroll
    for (int off = 16; off > 0; off >>= 1) vs += __shfl_xor(vs, off, 32);
    float rstd = rsqrtf(vs * (1.0f / 64.0f) + EPS_LN);

    out[base + lane]      = d0 * rstd * gamma[lane]      + beta[lane];
    out[base + lane + 32] = d1 * rstd * gamma[lane + 32] + beta[lane + 32];
}

// ---------------------------------------------------------------- launch
extern "C" void kernel_launch(void* const* d_in, const int* in_sizes, int n_in,
                              void* d_out, int out_size, void* d_ws, size_t ws_size,
                              hipStream_t stream) {
    const float* x         = (const float*)d_in[0];
    const int*   ei        = (const int*)  d_in[1];
    const float* edge_attr = (const float*)d_in[2];
    const float* W_node    = (const float*)d_in[3];
    const float* W_edge    = (const float*)d_in[4];
    const float* attn_src  = (const float*)d_in[5];
    const float* attn_dst  = (const float*)d_in[6];
    const float* gamma     = (const float*)d_in[7];
    const float* beta      = (const float*)d_in[8];
    float* out = (float*)d_out;

    int  n = in_sizes[0] / IN_DIM;
    long e = (long)in_sizes[2] / EDGE_DIM;
    const int* srcIdx = ei;
    const int* dstIdx = ei + e;

    // workspace layout (floats)
    float* ws    = (float*)d_ws;
    float* h     = ws;  ws += (long)n * OUT_DIM;
    float* a_src = ws;  ws += (long)n * H_HEADS;
    float* a_dst = ws;  ws += (long)n * H_HEADS;
    float* attn  = ws;  ws += e * H_HEADS;
    float* acc   = ws;  ws += (long)n * OUT_DIM;   // zeroed (contiguous with:)
    float* amax  = ws;  ws += (long)n * H_HEADS;   // zeroed
    float* asum  = ws;  ws += (long)n * H_HEADS;   // zeroed

    long zn = (long)n * (OUT_DIM + 2 * H_HEADS);   // acc|amax|asum contiguous
    zero_kernel<<<(int)((zn + 255) / 256), 256, 0, stream>>>(acc, zn);

    int n_tiles = (n + 15) / 16;
    node_gemm_wmma<<<n_tiles, 128, 0, stream>>>(x, W_node, h, n);

    node_proj<<<(n * H_HEADS + 255) / 256, 256, 0, stream>>>(
        h, attn_src, attn_dst, a_src, a_dst, n);

    edge_score<<<(int)((e + 255) / 256), 256, 0, stream>>>(
        srcIdx, dstIdx, edge_attr, W_edge, a_src, a_dst, attn, (int*)amax, e);

    long eh = e * H_HEADS;
    edge_softmax<<<(int)((eh + 255) / 256), 256, 0, stream>>>(
        dstIdx, amax, attn, asum, eh);

    long total = e * OUT_DIM;
    scatter_msg<<<(int)((total + 255) / 256), 256, 0, stream>>>(
        srcIdx, dstIdx, h, attn, asum, acc, total);

    residual_ln<<<(n + 7) / 8, 256, 0, stream>>>(acc, h, gamma, beta, out, n);
}